// CounterfactualAnswerLoss_35089882808675
// MI455X (gfx1250) — compile-verified
//
#include <hip/hip_runtime.h>
#include <math.h>

// ---------------------------------------------------------------------------
// CounterfactualAnswerLoss for MI455X (gfx1250, wave32)
//
// Roofline: dominated by streaming p_z -> p_cf_z (131MB + 131MB ~ 11us at
// 23.3 TB/s). Each 32000-float row is staged in LDS (125KB of the WGP's
// 320KB) so we do exactly one HBM read and one HBM write per row; the
// softmax normalization factor is folded into the write pass.
// h_ans reductions use V_WMMA_F32_16X16X4_F32 (exact f32) for row sums.
// ---------------------------------------------------------------------------

typedef __attribute__((ext_vector_type(2))) float v2f;
typedef __attribute__((ext_vector_type(8))) float v8f;

#define BATCH 64
#define KMAX  16
#define VOCAB 32000
#define NROWS (BATCH * KMAX)
#define HDIM  4096
#define DVN   10
#define EPSF  1e-8f

// Native transcendentals: v_log_f32 is log2, v_exp_f32 is exp2.
#define DEV_LOG2(x) __builtin_amdgcn_logf(x)
#define DEV_EXP2(x) __builtin_amdgcn_exp2f(x)

// ------------------------- Threefry-2x32 (JAX PRNG) ------------------------
__device__ __forceinline__ unsigned rotl32(unsigned v, int n) {
  return (v << n) | (v >> (32 - n));
}

__device__ __forceinline__ void threefry2x32(unsigned k0, unsigned k1,
                                             unsigned x0, unsigned x1,
                                             unsigned& y0, unsigned& y1) {
  unsigned k2 = k0 ^ k1 ^ 0x1BD11BDAu;
  x0 += k0; x1 += k1;
#define TFR(r) { x0 += x1; x1 = rotl32(x1, r); x1 ^= x0; }
  TFR(13) TFR(15) TFR(26) TFR(6)
  x0 += k1; x1 += k2 + 1u;
  TFR(17) TFR(29) TFR(16) TFR(24)
  x0 += k2; x1 += k0 + 2u;
  TFR(13) TFR(15) TFR(26) TFR(6)
  x0 += k0; x1 += k1 + 3u;
  TFR(17) TFR(29) TFR(16) TFR(24)
  x0 += k1; x1 += k2 + 4u;
  TFR(13) TFR(15) TFR(26) TFR(6)
  x0 += k2; x1 += k0 + 5u;
#undef TFR
  y0 = x0; y1 = x1;
}

__device__ __forceinline__ float bits_to_u01(unsigned bits) {
  return __uint_as_float((bits >> 9) | 0x3f800000u) - 1.0f;
}

// JAX threefry_random_bits: counts iota(n) split into halves (x0=i, x1=i+n/2);
// out = concat(y0_half, y1_half). Element idx of an n-element uniform draw:
__device__ __forceinline__ float tf_uniform(unsigned k0, unsigned k1,
                                            int idx, int half) {
  unsigned y0, y1;
  if (idx < half) {
    threefry2x32(k0, k1, (unsigned)idx, (unsigned)(idx + half), y0, y1);
    return bits_to_u01(y0);
  }
  threefry2x32(k0, k1, (unsigned)(idx - half), (unsigned)idx, y0, y1);
  return bits_to_u01(y1);
}

// ------------------- Kernel 1: per-(b,slot) control table ------------------
// For each of the 1024 (b,slot) rows decide: copy from src slot (itau<0) or
// temperature-warp with inv-tau (itau>0). Writes to workspace.
__global__ void ctrl_kernel(const int* __restrict__ k_vals,
                            int* __restrict__ srcArr,
                            float* __restrict__ itauArr) {
  int b = threadIdx.x;
  if (b >= BATCH) return;

  // key(0) split into 3: counts 0..5 pairs (0,3),(1,4),(2,5)
  unsigned y00, y10, y01, y11, y02, y12;
  threefry2x32(0u, 0u, 0u, 3u, y00, y10);
  threefry2x32(0u, 0u, 1u, 4u, y01, y11);
  threefry2x32(0u, 0u, 2u, 5u, y02, y12);
  const unsigned kd0 = y00, kd1 = y01;   // k_dec
  const unsigned kp0 = y02, kp1 = y10;   // k_perm
  const unsigned kt0 = y11, kt1 = y12;   // k_tau

  const int k = k_vals[b];
  const float u_dec = tf_uniform(kd0, kd1, b, BATCH / 2);
  const bool do_perm = (k >= 2) && (u_dec < 0.5f);  // PROB_TABLE[k]=0.5 for k>=2

  float w[KMAX]; int idxs[KMAX]; float tau[KMAX];
  for (int s = 0; s < KMAX; ++s) {
    int j = b * KMAX + s;
    float up = tf_uniform(kp0, kp1, j, (BATCH * KMAX) / 2);
    float ut = tf_uniform(kt0, kt1, j, (BATCH * KMAX) / 2);
    tau[s]  = fmaxf(0.7f, 0.7f + ut * 0.9f);   // uniform(0.7, 1.6) w/ lax.max
    w[s]    = (s < k) ? up : __builtin_inff();  // inactive -> +inf
    idxs[s] = s;
  }
  // stable insertion sort == stable argsort (tail keeps order)
  for (int i = 1; i < KMAX; ++i) {
    float kw = w[i]; int ki = idxs[i]; int j = i - 1;
    while (j >= 0 && w[j] > kw) { w[j + 1] = w[j]; idxs[j + 1] = idxs[j]; --j; }
    w[j + 1] = kw; idxs[j + 1] = ki;
  }
  for (int s = 0; s < KMAX; ++s) {
    int g = b * KMAX + s;
    if (s >= k)       { srcArr[g] = s;       itauArr[g] = -1.0f; }
    else if (do_perm) { srcArr[g] = idxs[s]; itauArr[g] = -1.0f; }
    else              { srcArr[g] = s;       itauArr[g] = 1.0f / tau[s]; }
  }
}

// --------------- Kernel 2: p_cf_z stream (1 block per row) -----------------
// Copy mode: LDS-staged row copy. Warp mode: t = p^(1/tau) via native
// v_log_f32/v_exp_f32 (exp2(log2(p)*itau)), block-reduce sum, scale on the
// write pass. Output base is d_out+3 floats (byte 12): element 0 and the
// last 3 are stored scalar so the 7999 middle float4 stores are 16B-aligned.
__global__ __launch_bounds__(256) void pz_kernel(
    const float* __restrict__ p_z, const int* __restrict__ srcArr,
    const float* __restrict__ itauArr, float* __restrict__ out /* d_out+3 */) {
  __shared__ float buf[VOCAB];     // 125 KB of the WGP's 320 KB LDS
  __shared__ float red[256];
  const int row = blockIdx.x;
  const int tid = threadIdx.x;
  const int b = row >> 4;
  const int src = srcArr[row];
  const float itau = itauArr[row];
  const float4* in4 =
      reinterpret_cast<const float4*>(p_z + (size_t)(b * KMAX + src) * VOCAB);

  float scale = 1.0f;
  if (itau < 0.0f) {  // copy (uniform across block)
    for (int j = tid; j < VOCAB / 4; j += 256) {
      float4 v = in4[j];
      buf[4 * j + 0] = v.x; buf[4 * j + 1] = v.y;
      buf[4 * j + 2] = v.z; buf[4 * j + 3] = v.w;
    }
  } else {            // temperature warp
    float lsum = 0.0f;
    for (int j = tid; j < VOCAB / 4; j += 256) {
      __builtin_prefetch(in4 + j + 1024, 0, 1);  // global_prefetch_b8
      float4 v = in4[j];
      float t0 = DEV_EXP2(DEV_LOG2(fmaxf(v.x, EPSF)) * itau);
      float t1 = DEV_EXP2(DEV_LOG2(fmaxf(v.y, EPSF)) * itau);
      float t2 = DEV_EXP2(DEV_LOG2(fmaxf(v.z, EPSF)) * itau);
      float t3 = DEV_EXP2(DEV_LOG2(fmaxf(v.w, EPSF)) * itau);
      buf[4 * j + 0] = t0; buf[4 * j + 1] = t1;
      buf[4 * j + 2] = t2; buf[4 * j + 3] = t3;
      lsum += (t0 + t1) + (t2 + t3);
    }
    red[tid] = lsum;
    __syncthreads();
    for (int s = 128; s > 0; s >>= 1) {
      if (tid < s) red[tid] += red[tid + s];
      __syncthreads();
    }
    scale = 1.0f / red[0];
  }
  __syncthreads();

  float* orow = out + (size_t)row * VOCAB;
  for (int t = tid; t < (VOCAB - 4) / 4; t += 256) {  // 7999 aligned b128 stores
    int e = 1 + 4 * t;                                 // byte off 16+16t ✓
    float4 v;
    v.x = buf[e] * scale;     v.y = buf[e + 1] * scale;
    v.z = buf[e + 2] * scale; v.w = buf[e + 3] * scale;
    *reinterpret_cast<float4*>(orow + e) = v;
  }
  if (tid == 0) orow[0] = buf[0] * scale;
  if (tid < 3) orow[VOCAB - 3 + tid] = buf[VOCAB - 3 + tid] * scale;
}

// ---------------- Kernel 3: digit JS loss -> d_out[0] ----------------------
__global__ void js_kernel(const float* __restrict__ lr,
                          const float* __restrict__ lc,
                          float* __restrict__ out) {
  __shared__ float red[BATCH];
  int b = threadIdx.x;
  const float* r = lr + b * DVN;
  const float* c = lc + b * DVN;
  float mr = -__builtin_inff(), mc = -__builtin_inff();
  for (int i = 0; i < DVN; ++i) { mr = fmaxf(mr, r[i]); mc = fmaxf(mc, c[i]); }
  float pr[DVN], pc[DVN], sr = 0.0f, sc = 0.0f;
  for (int i = 0; i < DVN; ++i) {
    pr[i] = expf(r[i] - mr); sr += pr[i];
    pc[i] = expf(c[i] - mc); sc += pc[i];
  }
  float klpm = 0.0f, klqm = 0.0f;
  for (int i = 0; i < DVN; ++i) {
    float p = fmaxf(pr[i] / sr, EPSF);
    float q = fmaxf(pc[i] / sc, EPSF);
    float lm = logf(0.5f * (p + q));
    klpm += p * (logf(p) - lm);
    klqm += q * (logf(q) - lm);
  }
  red[b] = 0.5f * (klpm + klqm);
  __syncthreads();
  if (b == 0) {
    float s = 0.0f;
    for (int i = 0; i < BATCH; ++i) s += red[i];
    out[0] = 0.69314718056f - s / (float)BATCH;  // log(2) - mean JS
  }
}

// ------ Kernel 4: h_ans reductions via V_WMMA_F32_16X16X4_F32 --------------
// 4 waves, one 16-row tile each. Each lane accumulates 2 f32 partials per
// row in exactly the 16x4 A-matrix layout (lanes 0-15: K0/K1 = cols 0,1 mod
// 4; lanes 16-31: K2/K3 = cols 2,3 mod 4). One WMMA against an all-ones B
// then yields exact f32 row sums (rows 0-7 on lane 0, rows 8-15 on lane 16).
__global__ __launch_bounds__(128) void hans_kernel(
    const float* __restrict__ hr, const float* __restrict__ hc,
    float* __restrict__ out) {
  __shared__ float cosv[BATCH];
  __shared__ float depv[BATCH];
  const int tid = threadIdx.x;
  const int wave = tid >> 5;
  const int lane = tid & 31;
  const int half = lane >> 4;
  const int row = wave * 16 + (lane & 15);
  const float2* rp = reinterpret_cast<const float2*>(hr + (size_t)row * HDIM);
  const float2* cp = reinterpret_cast<const float2*>(hc + (size_t)row * HDIM);

  float rc0 = 0, rc1 = 0, rr0 = 0, rr1 = 0, cc0 = 0, cc1 = 0, dd0 = 0, dd1 = 0;
#pragma unroll 4
  for (int j = 0; j < HDIM / 4; ++j) {
    float2 rv = rp[2 * j + half];   // cols 4j+2*half, +1 (8B-aligned loads)
    float2 cv = cp[2 * j + half];
    rc0 = fmaf(rv.x, cv.x, rc0); rc1 = fmaf(rv.y, cv.y, rc1);
    rr0 = fmaf(rv.x, rv.x, rr0); rr1 = fmaf(rv.y, rv.y, rr1);
    cc0 = fmaf(cv.x, cv.x, cc0); cc1 = fmaf(cv.y, cv.y, cc1);
    float dx = rv.x - cv.x, dy = rv.y - cv.y;
    dd0 = fmaf(dx, dx, dd0); dd1 = fmaf(dy, dy, dd1);
  }

  v2f ones; ones[0] = 1.0f; ones[1] = 1.0f;
  v8f cz = {0.f, 0.f, 0.f, 0.f, 0.f, 0.f, 0.f, 0.f};
  v2f a;
  a[0] = rc0; a[1] = rc1;
  v8f d_rc = __builtin_amdgcn_wmma_f32_16x16x4_f32(false, a, false, ones,
                                                   (short)0, cz, false, false);
  a[0] = rr0; a[1] = rr1;
  v8f d_rr = __builtin_amdgcn_wmma_f32_16x16x4_f32(false, a, false, ones,
                                                   (short)0, cz, false, false);
  a[0] = cc0; a[1] = cc1;
  v8f d_cc = __builtin_amdgcn_wmma_f32_16x16x4_f32(false, a, false, ones,
                                                   (short)0, cz, false, false);
  a[0] = dd0; a[1] = dd1;
  v8f d_dd = __builtin_amdgcn_wmma_f32_16x16x4_f32(false, a, false, ones,
                                                   (short)0, cz, false, false);

  if ((lane & 15) == 0) {   // lanes 0 (rows 0-7) and 16 (rows 8-15): col N=0
    int rbase = wave * 16 + half * 8;
#pragma unroll
    for (int r = 0; r < 8; ++r) {
      float rc = d_rc[r], rr = d_rr[r], cc = d_cc[r], dd = d_dd[r];
      float denom = fmaxf(sqrtf(rr), 1e-12f) * fmaxf(sqrtf(cc), 1e-12f);
      cosv[rbase + r] = rc / denom;
      depv[rbase + r] = sqrtf(dd);
    }
  }
  __syncthreads();
  if (tid == 0) {
    float sc = 0.0f, sd = 0.0f;
    for (int i = 0; i < BATCH; ++i) { sc += cosv[i]; sd += depv[i]; }
    out[1] = sc / (float)BATCH - 1.0f;  // loss_dep = mean cos - 1
    out[2] = sd / (float)BATCH;         // dep_norm
  }
}

// ---------------------------------------------------------------------------
extern "C" void kernel_launch(void* const* d_in, const int* in_sizes, int n_in,
                              void* d_out, int out_size, void* d_ws,
                              size_t ws_size, hipStream_t stream) {
  (void)in_sizes; (void)n_in; (void)out_size; (void)ws_size;
  const float* p_z = (const float*)d_in[0];
  const float* dlr = (const float*)d_in[1];
  const float* dlc = (const float*)d_in[2];
  const float* hr  = (const float*)d_in[3];
  const float* hc  = (const float*)d_in[4];
  const int*   kv  = (const int*)d_in[5];
  float* out = (float*)d_out;

  int*   ws_src  = (int*)d_ws;
  float* ws_itau = (float*)((char*)d_ws + NROWS * sizeof(int));

  hipLaunchKernelGGL(ctrl_kernel, dim3(1), dim3(64), 0, stream,
                     kv, ws_src, ws_itau);
  hipLaunchKernelGGL(pz_kernel, dim3(NROWS), dim3(256), 0, stream,
                     p_z, ws_src, ws_itau, out + 3);
  hipLaunchKernelGGL(js_kernel, dim3(1), dim3(BATCH), 0, stream,
                     dlr, dlc, out);
  hipLaunchKernelGGL(hans_kernel, dim3(1), dim3(128), 0, stream,
                     hr, hc, out);
}